// Top2Gating_33921651704035
// MI455X (gfx1250) — compile-verified
//
#include <hip/hip_runtime.h>
#include <hip/hip_bf16.h>
#include <math.h>

// ---------------------------------------------------------------------------
// Top2Gating for MI455X (gfx1250, wave32).
// Sizes (hardcoded per reference): G=8, S=2048, M=1024, E=64, C=64.
// ---------------------------------------------------------------------------

#define G_DIM 8
#define S_DIM 2048
#define M_DIM 1024
#define E_DIM 64
#define C_DIM 64
#define T_DIM (G_DIM * S_DIM)          // 16384 tokens
#define EC_DIM (E_DIM * C_DIM)         // 4096
#define COMBINE_ELEMS ((size_t)T_DIM * EC_DIM)   // 67,108,864

typedef __attribute__((ext_vector_type(2))) float v2f;
typedef __attribute__((ext_vector_type(4))) float v4f;
typedef __attribute__((ext_vector_type(8))) float v8f;

// ---------------------------------------------------------------------------
// Kernel A: logits[T,E] = inputs[T,M] x weight[M,E] via V_WMMA_F32_16X16X4_F32
// grid: T/16 blocks, 128 threads (4 waves). Wave w computes expert tile w.
// A-frag layout (ISA 7.12.2, 32-bit A 16x4): lanes 0-15 hold K=0,1 (v0,v1);
// lanes 16-31 hold K=2,3. B-frag mirrored (4x16, rows striped across lanes).
// ---------------------------------------------------------------------------
__global__ __launch_bounds__(128)
void gating_gemm_wmma(const float* __restrict__ A,    // [T, M]
                      const float* __restrict__ W,    // [M, E]
                      float* __restrict__ logits)     // [T, E]
{
    const int wave = threadIdx.x >> 5;      // expert tile 0..3
    const int lane = threadIdx.x & 31;
    const int tb   = blockIdx.x * 16;       // token tile base
    const int eb   = wave * 16;             // expert tile base
    const int half = lane >> 4;             // 0: K even pair, 1: K odd pair
    const int l    = lane & 15;

    const float* arow = A + (size_t)(tb + l) * M_DIM + 2 * half;   // A[tb+l][k + 2*half]
    const float* wcol = W + (size_t)(2 * half) * E_DIM + eb + l;   // W[k + 2*half][eb+l]

    v8f c = {};
    #pragma unroll 8
    for (int k = 0; k < M_DIM; k += 4) {
        // A fragment: two consecutive K values for this lane's row -> b64 load
        const float2 av = *reinterpret_cast<const float2*>(arow + k);
        v2f a; a.x = av.x; a.y = av.y;
        // B fragment: two K-rows of the weight, same expert column
        v2f b;
        b.x = wcol[(size_t)k * E_DIM];
        b.y = wcol[(size_t)(k + 1) * E_DIM];
        c = __builtin_amdgcn_wmma_f32_16x16x4_f32(false, a, false, b,
                                                  (short)0, c, false, false);
    }

    // C/D layout: VGPR v -> token row (v + 8*half), column l
    float* out = logits + (size_t)(tb + 8 * half) * E_DIM + eb + l;
    #pragma unroll
    for (int v = 0; v < 8; ++v)
        out[(size_t)v * E_DIM] = c[v];
}

// ---------------------------------------------------------------------------
// Kernel B: per-token softmax over E=64, top-1/top-2 selection.
// Overwrites logits with softmax probs (needed for aux-loss proxy).
// One thread per token. Also zeroes the aux-loss output slot.
// ---------------------------------------------------------------------------
__global__ __launch_bounds__(256)
void softmax_top2(float* __restrict__ probs,          // in: logits, out: softmax
                  int* __restrict__ idx1, int* __restrict__ idx2,
                  float* __restrict__ g1raw, float* __restrict__ g2raw,
                  float* __restrict__ aux_out)
{
    const int t = blockIdx.x * blockDim.x + threadIdx.x;
    if (t == 0) *aux_out = 0.0f;     // stream-ordered before kernel C atomics
    if (t >= T_DIM) return;

    float* row = probs + (size_t)t * E_DIM;
    float p[E_DIM];
    float mx = -INFINITY;
    #pragma unroll
    for (int e = 0; e < E_DIM; ++e) { p[e] = row[e]; mx = fmaxf(mx, p[e]); }
    float s = 0.0f;
    #pragma unroll
    for (int e = 0; e < E_DIM; ++e) { p[e] = __expf(p[e] - mx); s += p[e]; }
    const float inv = 1.0f / s;

    // argmax (first max wins -> strict greater-than, ascending scan)
    int   i1 = 0; float b1 = -1.0f;
    #pragma unroll
    for (int e = 0; e < E_DIM; ++e) { if (p[e] > b1) { b1 = p[e]; i1 = e; } }
    int   i2 = 0; float b2 = -1.0f;
    #pragma unroll
    for (int e = 0; e < E_DIM; ++e) {
        float v = (e == i1) ? -1.0f : p[e];
        if (v > b2) { b2 = v; i2 = e; }
    }

    #pragma unroll
    for (int e = 0; e < E_DIM; ++e) row[e] = p[e] * inv;

    idx1[t]  = i1;
    idx2[t]  = i2;
    g1raw[t] = b1 * inv;
    g2raw[t] = b2 * inv;
}

// ---------------------------------------------------------------------------
// Kernel C: per-group exclusive-cumsum positions (serial over S per expert),
// capacity logic, and aux loss. grid = G blocks, 64 threads (1 per expert).
// idx arrays staged through LDS in 64-token chunks.
// ---------------------------------------------------------------------------
__global__ __launch_bounds__(64)
void scan_positions_aux(const float* __restrict__ probs,
                        const int* __restrict__ idx1, const int* __restrict__ idx2,
                        int* __restrict__ pos1, int* __restrict__ pos2,
                        float* __restrict__ aux_out)
{
    __shared__ int   sIdx[64];
    __shared__ float sRed[64];
    const int g = blockIdx.x;
    const int e = threadIdx.x;
    const int base = g * S_DIM;

    // ---- phase 1: top-1 positions ----
    int cnt1 = 0;
    for (int chunk = 0; chunk < S_DIM; chunk += 64) {
        __syncthreads();
        sIdx[e] = idx1[base + chunk + e];
        __syncthreads();
        #pragma unroll 8
        for (int j = 0; j < 64; ++j) {
            if (sIdx[j] == e) { pos1[base + chunk + j] = cnt1; ++cnt1; }
        }
    }
    const int total1 = cnt1;                  // pre-drop count (for aux)
    const int kept1  = min(cnt1, C_DIM);      // mask_1_count after capacity drop

    // ---- phase 2: top-2 positions (offset by kept top-1 count) ----
    int cnt2 = 0;
    for (int chunk = 0; chunk < S_DIM; chunk += 64) {
        __syncthreads();
        sIdx[e] = idx2[base + chunk + e];
        __syncthreads();
        #pragma unroll 8
        for (int j = 0; j < 64; ++j) {
            if (sIdx[j] == e) { pos2[base + chunk + j] = kept1 + cnt2; ++cnt2; }
        }
    }

    // ---- phase 3: aux loss partial: density_1_proxy * density_1 ----
    float psum = 0.0f;
    for (int s = 0; s < S_DIM; ++s)
        psum += probs[(size_t)(base + s) * E_DIM + e];   // coalesced across e

    const float d       = 1.0f + 1e-6f;                  // mean(importance)+eps
    const float density = ((float)total1 / (float)S_DIM) / d;
    const float proxy   = (psum / (float)S_DIM) / d;
    sRed[e] = density * proxy;
    __syncthreads();
    for (int off = 32; off > 0; off >>= 1) {
        if (e < off) sRed[e] += sRed[e + off];
        __syncthreads();
    }
    // aux = E*E * mean over (G,E) = (E/G) * sum = 8 * sum
    if (e == 0) atomicAdd(aux_out, ((float)E_DIM / (float)G_DIM) * sRed[0]);
}

// ---------------------------------------------------------------------------
// Kernel D: fused zero + scatter of combine_tensor and dispatch_mask.
// One block per token, 256 threads x 16 floats each = the token's 4096-float
// slab in each output. 128-bit non-temporal streaming stores (512 MB > L2).
// ---------------------------------------------------------------------------
__global__ __launch_bounds__(256)
void write_outputs(const int* __restrict__ idx1, const int* __restrict__ idx2,
                   const float* __restrict__ g1raw, const float* __restrict__ g2raw,
                   const int* __restrict__ pos1, const int* __restrict__ pos2,
                   float* __restrict__ out)
{
    const int t   = blockIdx.x;
    const int tid = threadIdx.x;

    const int   e1 = idx1[t],  e2 = idx2[t];
    const int   p1 = pos1[t],  p2 = pos2[t];
    const bool  k1 = p1 < C_DIM, k2 = p2 < C_DIM;
    float g1 = k1 ? g1raw[t] : 0.0f;
    float g2 = k2 ? g2raw[t] : 0.0f;
    float denom = g1 + g2;
    denom = denom > 0.0f ? denom : 1.0f;
    const float c1 = g1 / denom;
    const float c2 = g2 / denom;
    const int   o1 = e1 * C_DIM + p1;     // flat (E,C) offset for top-1
    const int   o2 = e2 * C_DIM + p2;     // flat (E,C) offset for top-2

    float* comb = out + (size_t)t * EC_DIM;
    float* disp = out + COMBINE_ELEMS + (size_t)t * EC_DIM;
    const int basei = tid * 16;

    float vals[16];
    #pragma unroll
    for (int i = 0; i < 16; ++i) {
        const int idx = basei + i;
        float v = 0.0f;
        if (k1 && idx == o1) v = c1;
        if (k2 && idx == o2) v = c2;
        vals[i] = v;
    }
    #pragma unroll
    for (int q = 0; q < 4; ++q) {
        v4f cv, dv;
        #pragma unroll
        for (int i = 0; i < 4; ++i) {
            const float v = vals[4 * q + i];
            cv[i] = v;
            dv[i] = (v != 0.0f) ? 1.0f : 0.0f;
        }
        __builtin_nontemporal_store(cv, reinterpret_cast<v4f*>(comb + basei + 4 * q));
        __builtin_nontemporal_store(dv, reinterpret_cast<v4f*>(disp + basei + 4 * q));
    }
}

// ---------------------------------------------------------------------------
// Host-side launcher
// ---------------------------------------------------------------------------
extern "C" void kernel_launch(void* const* d_in, const int* in_sizes, int n_in,
                              void* d_out, int out_size, void* d_ws, size_t ws_size,
                              hipStream_t stream)
{
    (void)in_sizes; (void)n_in; (void)out_size; (void)ws_size;

    const float* inputs = (const float*)d_in[0];   // [G,S,M]
    const float* weight = (const float*)d_in[1];   // [M,E]

    // workspace layout (~4.4 MB)
    float* probs = (float*)d_ws;                                   // T*E f32
    int*   idx1  = (int*)((char*)d_ws + (size_t)T_DIM * E_DIM * 4);
    int*   idx2  = idx1 + T_DIM;
    float* g1raw = (float*)(idx2 + T_DIM);
    float* g2raw = g1raw + T_DIM;
    int*   pos1  = (int*)(g2raw + T_DIM);
    int*   pos2  = pos1 + T_DIM;

    float* out = (float*)d_out;
    float* aux = out + 2 * COMBINE_ELEMS;          // scalar aux loss slot

    // A: logits via WMMA
    gating_gemm_wmma<<<T_DIM / 16, 128, 0, stream>>>(inputs, weight, probs);
    // B: softmax + top-2 (+ zero aux slot)
    softmax_top2<<<T_DIM / 256, 256, 0, stream>>>(probs, idx1, idx2, g1raw, g2raw, aux);
    // C: per-group position scans + aux loss
    scan_positions_aux<<<G_DIM, 64, 0, stream>>>(probs, idx1, idx2, pos1, pos2, aux);
    // D: stream the 512 MB dense outputs with inline scatter
    write_outputs<<<T_DIM, 256, 0, stream>>>(idx1, idx2, g1raw, g2raw, pos1, pos2, out);
}